// GrassmannAverageProjection_87436944212009
// MI455X (gfx1250) — compile-verified
//
#include <hip/hip_runtime.h>
#include <math.h>

// ---------------------------------------------------------------------------
// GrassmannAverageProjection for MI455X (gfx1250, wave32, WMMA)
//   IN_FRAMES=128, k=16, NUM_BLOCKS=8, D=3*224*224=150528
// Long-K (K=D) contractions -> v_wmma_f32_16x16x4_f32, unrolled x4 so each
//   wave has 4-16 loads in flight before the first s_wait_loadcnt.
// Short-K (K=16) row transforms -> VALU FMA, W matrices staged in LDS
// 16x16 dense LA (Cholesky / inverse / Jacobi eigen) -> single-thread kernels
// ---------------------------------------------------------------------------

#define GAP_D        150528
#define GAP_NB       8
#define RED_NS       64          // workgroups per D-contraction
#define RED_WAVES    8           // waves per workgroup (256 threads)
#define RED_TOTALW   (RED_NS * RED_WAVES)   // 512 waves
#define EPS_JITTER   0.01f

typedef float v2f __attribute__((ext_vector_type(2)));
typedef float v8f __attribute__((ext_vector_type(8)));

__device__ __forceinline__ v8f wmma_f32_k4(v2f a, v2f b, v8f c) {
  return __builtin_amdgcn_wmma_f32_16x16x4_f32(false, a, false, b, (short)0, c,
                                               false, false);
}

// -------------------- cross-wave 16x16 reduction helper ---------------------
__device__ __forceinline__ void reduce_and_atomic(v8f c, int tid, int wid,
                                                  int hh, int mn,
                                                  float* __restrict__ dst) {
  __shared__ float red[RED_WAVES * 256];
#pragma unroll
  for (int r = 0; r < 8; ++r)
    red[wid * 256 + (r + 8 * hh) * 16 + mn] = c[r];
  __syncthreads();
  float s = 0.f;
#pragma unroll
  for (int w = 0; w < RED_WAVES; ++w) s += red[w * 256 + tid];
  atomicAdd(&dst[tid], s);
}

// -------------------- Gram: acc += Xr * Xr^T, Xr is [16, D] row-major -------
__global__ __launch_bounds__(256) void gap_gram16(const float* __restrict__ Xr,
                                                  float* __restrict__ acc) {
  const int tid = threadIdx.x;
  const int lane = tid & 31, wid = tid >> 5;
  const int hh = lane >> 4, mn = lane & 15;
  const float* rowp = Xr + (size_t)mn * GAP_D + 2 * hh;

  v8f c0, c1;
#pragma unroll
  for (int r = 0; r < 8; ++r) { c0[r] = 0.f; c1[r] = 0.f; }

  const int gw = blockIdx.x * RED_WAVES + wid;
  for (int k = gw * 16; k < GAP_D; k += RED_TOTALW * 16) {
    v2f a[4];
#pragma unroll
    for (int u = 0; u < 4; ++u)
      a[u] = *reinterpret_cast<const v2f*>(rowp + k + 4 * u);
    c0 = wmma_f32_k4(a[0], a[0], c0);
    c1 = wmma_f32_k4(a[1], a[1], c1);
    c0 = wmma_f32_k4(a[2], a[2], c0);
    c1 = wmma_f32_k4(a[3], a[3], c1);
  }
  reduce_and_atomic(c0 + c1, tid, wid, hh, mn, acc);
}

// -------------------- acc += Mc^T * Xc, both [D,16] row-major ---------------
__global__ __launch_bounds__(256) void gap_crossgram(
    const float* __restrict__ Mc, const float* __restrict__ Xc,
    float* __restrict__ acc) {
  const int tid = threadIdx.x;
  const int lane = tid & 31, wid = tid >> 5;
  const int hh = lane >> 4, mn = lane & 15;

  v8f c0, c1;
#pragma unroll
  for (int r = 0; r < 8; ++r) { c0[r] = 0.f; c1[r] = 0.f; }

  const int gw = blockIdx.x * RED_WAVES + wid;
  for (int k = gw * 16; k < GAP_D; k += RED_TOTALW * 16) {
    v2f a[4], b[4];
#pragma unroll
    for (int u = 0; u < 4; ++u) {
      const size_t k0 = (size_t)(k + 4 * u + 2 * hh) * 16 + mn;
      a[u].x = Mc[k0];       // A[m][k] = Mc[k][m]
      a[u].y = Mc[k0 + 16];
      b[u].x = Xc[k0];       // B[k][n] = Xc[k][n]
      b[u].y = Xc[k0 + 16];
    }
    c0 = wmma_f32_k4(a[0], b[0], c0);
    c1 = wmma_f32_k4(a[1], b[1], c1);
    c0 = wmma_f32_k4(a[2], b[2], c0);
    c1 = wmma_f32_k4(a[3], b[3], c1);
  }
  reduce_and_atomic(c0 + c1, tid, wid, hh, mn, acc);
}

// -------------------- proj: out[tile] += x_rows * fm ------------------------
__global__ __launch_bounds__(256) void gap_proj(const float* __restrict__ x,
                                                const float* __restrict__ fm,
                                                float* __restrict__ out) {
  const int tid = threadIdx.x;
  const int lane = tid & 31, wid = tid >> 5;
  const int hh = lane >> 4, mn = lane & 15;
  const int m0 = blockIdx.y * 16;
  const float* rowp = x + (size_t)(m0 + mn) * GAP_D + 2 * hh;

  v8f c0, c1;
#pragma unroll
  for (int r = 0; r < 8; ++r) { c0[r] = 0.f; c1[r] = 0.f; }

  const int nwaves = gridDim.x * RED_WAVES;
  const int gw = blockIdx.x * RED_WAVES + wid;
  for (int k = gw * 16; k < GAP_D; k += nwaves * 16) {
    v2f a[4], b[4];
#pragma unroll
    for (int u = 0; u < 4; ++u) {
      a[u] = *reinterpret_cast<const v2f*>(rowp + k + 4 * u);
      const size_t k0 = (size_t)(k + 4 * u + 2 * hh) * 16 + mn;
      b[u].x = fm[k0];
      b[u].y = fm[k0 + 16];
    }
    c0 = wmma_f32_k4(a[0], b[0], c0);
    c1 = wmma_f32_k4(a[1], b[1], c1);
    c0 = wmma_f32_k4(a[2], b[2], c0);
    c1 = wmma_f32_k4(a[3], b[3], c1);
  }
  // element (m,n) of tile maps to out[(m0+m)*16+n] = out[m0*16 + tid]
  reduce_and_atomic(c0 + c1, tid, wid, hh, mn, out + (size_t)m0 * 16);
}

// -------------------- Y[d][:] = Arow(d)*W1 (+ Brow(d)*W2) -------------------
// A element (d,j) at A[d*sad + j*saj]; B (nullable) is [D,16] row-major.
__global__ __launch_bounds__(256) void gap_apply_right(
    const float* __restrict__ A, long sad, long saj,
    const float* __restrict__ B, const float* __restrict__ W1g,
    const float* __restrict__ W2g, float* __restrict__ Y) {
  __shared__ float W1[256], W2[256];
  const int t = threadIdx.x;
  W1[t] = W1g[t];
  W2[t] = (B != nullptr) ? W2g[t] : 0.f;
  __syncthreads();

  const size_t d = (size_t)blockIdx.x * 256 + t;
  const float* ap = A + (size_t)d * (size_t)sad;
  float a[16], y[16];
#pragma unroll
  for (int j = 0; j < 16; ++j) a[j] = ap[(size_t)j * (size_t)saj];
#pragma unroll
  for (int cc = 0; cc < 16; ++cc) y[cc] = 0.f;
#pragma unroll
  for (int r = 0; r < 16; ++r) {
    const float av = a[r];
#pragma unroll
    for (int cc = 0; cc < 16; ++cc) y[cc] = fmaf(av, W1[r * 16 + cc], y[cc]);
  }
  if (B != nullptr) {
    const float* bp = B + d * 16;
#pragma unroll
    for (int r = 0; r < 16; ++r) {
      const float bv = bp[r];
#pragma unroll
      for (int cc = 0; cc < 16; ++cc) y[cc] = fmaf(bv, W2[r * 16 + cc], y[cc]);
    }
  }
  float* yp = Y + d * 16;
#pragma unroll
  for (int j = 0; j < 16; ++j) yp[j] = y[j];
}

// -------------------- small helpers -----------------------------------------
__global__ void gap_zero(float* __restrict__ p, int n) {
  int i = blockIdx.x * blockDim.x + threadIdx.x;
  if (i < n) p[i] = 0.f;
}

__global__ void gap_penalty(const float* __restrict__ w, float* __restrict__ out) {
  float ref = 0.f;
  for (int n = 2; n < GAP_NB + 2; ++n) ref += 1.0f / (float)n;
  float s = 0.f;
  for (int i = 0; i < GAP_NB; ++i) s += fabsf(w[i]);
  const float d = ref - s;
  out[0] = d * d;
}

// Cholesky of (acc + eps I), invert lower-tri, emit W1 = L^{-T}
__global__ void gap_chol(const float* __restrict__ acc, float* __restrict__ W1) {
  __shared__ float Lm[256], Li[256];
  for (int i = 0; i < 16; ++i)
    for (int j = 0; j < 16; ++j)
      Lm[i * 16 + j] = acc[i * 16 + j] + (i == j ? EPS_JITTER : 0.f);
  for (int j = 0; j < 16; ++j) {
    float s = Lm[j * 17];
    for (int k = 0; k < j; ++k) s -= Lm[j * 16 + k] * Lm[j * 16 + k];
    const float dj = sqrtf(fmaxf(s, 1e-30f));
    Lm[j * 17] = dj;
    for (int i = j + 1; i < 16; ++i) {
      float s2 = Lm[i * 16 + j];
      for (int k = 0; k < j; ++k) s2 -= Lm[i * 16 + k] * Lm[j * 16 + k];
      Lm[i * 16 + j] = s2 / dj;
    }
  }
  for (int i = 0; i < 256; ++i) Li[i] = 0.f;
  for (int j = 0; j < 16; ++j) {
    Li[j * 17] = 1.f / Lm[j * 17];
    for (int i = j + 1; i < 16; ++i) {
      float s = 0.f;
      for (int k = j; k < i; ++k) s += Lm[i * 16 + k] * Li[k * 16 + j];
      Li[i * 16 + j] = -s / Lm[i * 17];
    }
  }
  for (int r = 0; r < 16; ++r)
    for (int c = 0; c < 16; ++c) W1[r * 16 + c] = Li[c * 16 + r];  // L^{-T}
}

// Gauss-Jordan inverse with partial pivoting; W1 = A^{-1}, W2 = -I
__global__ void gap_inv(const float* __restrict__ acc, float* __restrict__ W1,
                        float* __restrict__ W2) {
  __shared__ float M[16][32];
  for (int i = 0; i < 16; ++i)
    for (int j = 0; j < 16; ++j) {
      M[i][j] = acc[i * 16 + j];
      M[i][16 + j] = (i == j) ? 1.f : 0.f;
    }
  for (int col = 0; col < 16; ++col) {
    int piv = col;
    float best = fabsf(M[col][col]);
    for (int r = col + 1; r < 16; ++r)
      if (fabsf(M[r][col]) > best) { best = fabsf(M[r][col]); piv = r; }
    if (piv != col)
      for (int j = 0; j < 32; ++j) {
        const float t = M[col][j]; M[col][j] = M[piv][j]; M[piv][j] = t;
      }
    float dv = M[col][col];
    if (fabsf(dv) < 1e-30f) dv = 1e-30f;
    const float inv = 1.f / dv;
    for (int j = 0; j < 32; ++j) M[col][j] *= inv;
    for (int r = 0; r < 16; ++r) {
      if (r == col) continue;
      const float f = M[r][col];
      for (int j = 0; j < 32; ++j) M[r][j] -= f * M[col][j];
    }
  }
  for (int i = 0; i < 16; ++i)
    for (int j = 0; j < 16; ++j) {
      W1[i * 16 + j] = M[i][16 + j];
      W2[i * 16 + j] = (i == j) ? -1.f : 0.f;
    }
}

// cyclic Jacobi eigen, 16x16 symmetric; A destroyed (diag -> eigvals), V cols
__device__ void jacobi16(float* A, float* V) {
  for (int i = 0; i < 16; ++i)
    for (int j = 0; j < 16; ++j) V[i * 16 + j] = (i == j) ? 1.f : 0.f;
  for (int sweep = 0; sweep < 30; ++sweep) {
    for (int p = 0; p < 15; ++p)
      for (int q = p + 1; q < 16; ++q) {
        const float apq = A[p * 16 + q];
        if (fabsf(apq) < 1e-20f) continue;
        const float app = A[p * 17], aqq = A[q * 17];
        const float tau = (aqq - app) / (2.f * apq);
        const float t =
            (tau >= 0.f ? 1.f : -1.f) / (fabsf(tau) + sqrtf(1.f + tau * tau));
        const float cs = 1.f / sqrtf(1.f + t * t);
        const float sn = t * cs;
        for (int k = 0; k < 16; ++k) {
          const float akp = A[k * 16 + p], akq = A[k * 16 + q];
          A[k * 16 + p] = cs * akp - sn * akq;
          A[k * 16 + q] = sn * akp + cs * akq;
        }
        for (int k = 0; k < 16; ++k) {
          const float apk = A[p * 16 + k], aqk = A[q * 16 + k];
          A[p * 16 + k] = cs * apk - sn * aqk;
          A[q * 16 + k] = sn * apk + cs * aqk;
        }
        for (int k = 0; k < 16; ++k) {
          const float vkp = V[k * 16 + p], vkq = V[k * 16 + q];
          V[k * 16 + p] = cs * vkp - sn * vkq;
          V[k * 16 + q] = sn * vkp + cs * vkq;
        }
      }
  }
}

// from G = P^T P and weight w: W1 = Wc, W2 = Wlog*Wsinc  (log+exp collapsed)
__global__ void gap_eig(const float* __restrict__ acc,
                        const float* __restrict__ wts, int wi,
                        float* __restrict__ W1, float* __restrict__ W2) {
  __shared__ float G[256], Aw[256], V[256], Wlog[256], T[256], Wsc[256];
  const float w = wts[wi];
  for (int i = 0; i < 256; ++i) { G[i] = acc[i]; Aw[i] = acc[i]; }
  jacobi16(Aw, V);
  float f[16];
  for (int i = 0; i < 16; ++i) {
    const float s = sqrtf(fmaxf(Aw[i * 17], 0.f));
    f[i] = w * (s > 1e-8f ? atanf(s) / s : 1.f);   // w * atan(s)/s
  }
  for (int a = 0; a < 16; ++a)
    for (int b = 0; b < 16; ++b) {
      float s = 0.f;
      for (int i = 0; i < 16; ++i) s += V[a * 16 + i] * f[i] * V[b * 16 + i];
      Wlog[a * 16 + b] = s;
    }
  // Gh = Wlog^T G Wlog  (H = P*Wlog, Gh = H^T H)
  for (int a = 0; a < 16; ++a)
    for (int b = 0; b < 16; ++b) {
      float s = 0.f;
      for (int c = 0; c < 16; ++c) s += G[a * 16 + c] * Wlog[c * 16 + b];
      T[a * 16 + b] = s;
    }
  for (int a = 0; a < 16; ++a)
    for (int b = 0; b < 16; ++b) {
      float s = 0.f;
      for (int c = 0; c < 16; ++c) s += Wlog[c * 16 + a] * T[c * 16 + b];
      Aw[a * 16 + b] = s;
    }
  jacobi16(Aw, V);
  float cv[16], sv[16];
  for (int i = 0; i < 16; ++i) {
    const float s = sqrtf(fmaxf(Aw[i * 17], 0.f));
    cv[i] = cosf(s);
    sv[i] = (s > 1e-8f ? sinf(s) / s : 1.f);
  }
  for (int a = 0; a < 16; ++a)
    for (int b = 0; b < 16; ++b) {
      float sc = 0.f, ss = 0.f;
      for (int i = 0; i < 16; ++i) {
        const float vv = V[a * 16 + i] * V[b * 16 + i];
        sc += vv * cv[i];
        ss += vv * sv[i];
      }
      W1[a * 16 + b] = sc;        // Wc
      Wsc[a * 16 + b] = ss;       // Wsinc
    }
  for (int a = 0; a < 16; ++a)
    for (int b = 0; b < 16; ++b) {
      float s = 0.f;
      for (int c = 0; c < 16; ++c) s += Wlog[a * 16 + c] * Wsc[c * 16 + b];
      W2[a * 16 + b] = s;         // Wp
    }
}

// ---------------------------------------------------------------------------
extern "C" void kernel_launch(void* const* d_in, const int* in_sizes, int n_in,
                              void* d_out, int out_size, void* d_ws,
                              size_t ws_size, hipStream_t stream) {
  (void)in_sizes; (void)n_in; (void)out_size; (void)ws_size;
  const float* x = (const float*)d_in[0];      // [128, D]
  const float* w = (const float*)d_in[1];      // [8]
  float* out = (float*)d_out;                  // [128*16] proj + [1] penalty
  float* ws = (float*)d_ws;

  float* ACC = ws;                             // 16x16 contraction accumulator
  float* W1 = ws + 256;
  float* W2 = ws + 512;
  const size_t DN = (size_t)GAP_D * 16;
  float* M  = ws + 1024;                       // [D,16] ping
  float* Mn = ws + 1024 + DN;                  //        pong
  float* P  = ws + 1024 + 2 * DN;              // tangent precursor
  float* X  = ws + 1024 + 3 * DN;              // current orth block

  const dim3 blk(256);
  const int applyGrid = GAP_D / 256;           // 588

  auto orthogonalize = [&](int b, float* dst) {
    gap_zero<<<1, 256, 0, stream>>>(ACC, 256);
    gap_gram16<<<RED_NS, blk, 0, stream>>>(x + (size_t)b * 16 * GAP_D, ACC);
    gap_chol<<<1, 1, 0, stream>>>(ACC, W1);
    // U = block^T * L^{-T}: A(d,j) = x[(b*16+j)*D + d] -> sad=1, saj=D
    gap_apply_right<<<applyGrid, blk, 0, stream>>>(
        x + (size_t)b * 16 * GAP_D, 1L, (long)GAP_D, nullptr, W1, W2, dst);
  };

  orthogonalize(0, M);
  for (int i = 1; i < GAP_NB; ++i) {
    orthogonalize(i, X);
    // A = M^T X
    gap_zero<<<1, 256, 0, stream>>>(ACC, 256);
    gap_crossgram<<<RED_NS, blk, 0, stream>>>(M, X, ACC);
    gap_inv<<<1, 1, 0, stream>>>(ACC, W1, W2);
    // P = X * A^{-1} - M
    gap_apply_right<<<applyGrid, blk, 0, stream>>>(X, 16L, 1L, M, W1, W2, P);
    // G = P^T P
    gap_zero<<<1, 256, 0, stream>>>(ACC, 256);
    gap_crossgram<<<RED_NS, blk, 0, stream>>>(P, P, ACC);
    gap_eig<<<1, 1, 0, stream>>>(ACC, w, i, W1, W2);
    // M' = M*Wc + P*Wp
    gap_apply_right<<<applyGrid, blk, 0, stream>>>(M, 16L, 1L, P, W1, W2, Mn);
    float* t = M; M = Mn; Mn = t;
  }

  // proj = x @ fm  (8 row tiles of 16, WMMA reduction, atomic accumulate)
  gap_zero<<<8, 256, 0, stream>>>(out, 128 * 16);
  gap_proj<<<dim3(16, 8), blk, 0, stream>>>(x, M, out);
  gap_penalty<<<1, 1, 0, stream>>>(w, out + 128 * 16);
}